// models3_78941498900665
// MI455X (gfx1250) — compile-verified
//
#include <hip/hip_runtime.h>
#include <hip/hip_bf16.h>

#define BSZ 2048

typedef float v2f __attribute__((ext_vector_type(2)));
typedef float v8f __attribute__((ext_vector_type(8)));

__device__ __forceinline__ float selu_f(float x) {
    const float scale = 1.0507009873554805f, alpha = 1.6732632423543772f;
    return x > 0.f ? scale * x : scale * alpha * (expf(x) - 1.f);
}
__device__ __forceinline__ float sigmoid_f(float x) { return 1.f / (1.f + expf(-x)); }
__device__ __forceinline__ float gelu_f(float x) { return 0.5f * x * (1.f + erff(x * 0.70710678118654752f)); }

// ---------------- normalize_A: L = D^-1/2 relu(A) D^-1/2 ----------------
__global__ void k_normA(const float* __restrict__ A, float* __restrict__ Lm) {
    __shared__ float dinv[64];
    int t = threadIdx.x;
    if (t < 62) {
        float s = 0.f;
        for (int j = 0; j < 62; ++j) s += fmaxf(A[t * 62 + j], 0.f);
        dinv[t] = 1.f / sqrtf(s + 1e-10f);
    }
    __syncthreads();
    for (int idx = t; idx < 62 * 62; idx += blockDim.x) {
        int i = idx / 62, j = idx % 62;
        Lm[idx] = dinv[i] * fmaxf(A[idx], 0.f) * dinv[j];
    }
}

// ---------------- conv1d (valid) + BN(eval) + SELU, input = concat(X1,X2) along L ----------------
__global__ void k_conv(const float* __restrict__ X1, int L1,
                       const float* __restrict__ X2, int L2,
                       const float* __restrict__ W, const float* __restrict__ bias,
                       const float* __restrict__ bng, const float* __restrict__ bnb,
                       int KW, float* __restrict__ Y) {
    __shared__ float sx[62 * 14];
    int b = blockIdx.x;
    int Lin = L1 + L2;
    int Tout = Lin - KW + 1;
    for (int i = threadIdx.x; i < 62 * Lin; i += blockDim.x) {
        int c = i / Lin, l = i % Lin;
        sx[i] = (l < L1) ? X1[(b * 62 + c) * L1 + l] : X2[(b * 62 + c) * L2 + (l - L1)];
    }
    __syncthreads();
    const float bnscale = 1.f / sqrtf(1.f + 1e-5f);
    for (int idx = threadIdx.x; idx < 62 * Tout; idx += blockDim.x) {
        int o = idx / Tout, t = idx % Tout;
        float acc = bias[o];
        const float* wr = W + o * 62 * KW;
        for (int c = 0; c < 62; ++c)
            for (int k = 0; k < KW; ++k)
                acc += sx[c * Lin + t + k] * wr[c * KW + k];
        float y = acc * (bng[o] * bnscale) + bnb[o];
        Y[(b * 62 + o) * Tout + t] = selu_f(y);
    }
}

// ---------------- channel attention on concat(X1,X2,X3) along L ----------------
__global__ void k_chanatt(const float* __restrict__ X1, int L1,
                          const float* __restrict__ X2, int L2,
                          const float* __restrict__ X3, int L3,
                          const float* __restrict__ W1, const float* __restrict__ W2,
                          float* __restrict__ Y) {
    __shared__ float sx[62 * 30];
    __shared__ float avg[62], mx[62], hid_a[15], hid_m[15], gate[62];
    int b = blockIdx.x;
    int Lt = L1 + L2 + L3;
    for (int i = threadIdx.x; i < 62 * Lt; i += blockDim.x) {
        int c = i / Lt, l = i % Lt;
        float v;
        if (l < L1) v = X1[(b * 62 + c) * L1 + l];
        else if (l < L1 + L2) v = X2[(b * 62 + c) * L2 + (l - L1)];
        else v = X3[(b * 62 + c) * L3 + (l - L1 - L2)];
        sx[c * Lt + l] = v;
    }
    __syncthreads();
    if (threadIdx.x < 62) {
        int c = threadIdx.x;
        float s = 0.f, m = -INFINITY;
        for (int l = 0; l < Lt; ++l) { float v = sx[c * Lt + l]; s += v; m = fmaxf(m, v); }
        avg[c] = s / (float)Lt; mx[c] = m;
    }
    __syncthreads();
    if (threadIdx.x < 15) {
        int j = threadIdx.x;
        float sa = 0.f, sm2 = 0.f;
        for (int c = 0; c < 62; ++c) { float w = W1[j * 62 + c]; sa += avg[c] * w; sm2 += mx[c] * w; }
        hid_a[j] = fmaxf(sa, 0.f); hid_m[j] = fmaxf(sm2, 0.f);
    }
    __syncthreads();
    if (threadIdx.x < 62) {
        int c = threadIdx.x;
        float s = 0.f;
        for (int j = 0; j < 15; ++j) s += (hid_a[j] + hid_m[j]) * W2[c * 15 + j];
        gate[c] = sigmoid_f(s);
    }
    __syncthreads();
    for (int i = threadIdx.x; i < 62 * Lt; i += blockDim.x) {
        int c = i / Lt;
        Y[(size_t)b * 62 * Lt + i] = gate[c] * sx[i];
    }
}

// ---------------- layernorm over last dim 5 ----------------
__global__ void k_ln(const float* __restrict__ X, const float* __restrict__ g,
                     const float* __restrict__ bb, float* __restrict__ Y, int n_groups) {
    int idx = blockIdx.x * blockDim.x + threadIdx.x;
    if (idx >= n_groups) return;
    const float* xp = X + (size_t)idx * 5;
    float m = 0.f;
    for (int l = 0; l < 5; ++l) m += xp[l];
    m *= 0.2f;
    float v = 0.f;
    for (int l = 0; l < 5; ++l) { float d = xp[l] - m; v += d * d; }
    v *= 0.2f;
    float inv = rsqrtf(v + 1e-5f);
    for (int l = 0; l < 5; ++l) Y[(size_t)idx * 5 + l] = (xp[l] - m) * inv * g[l] + bb[l];
}

// ---------------- generic f32 WMMA GEMM: Y = act(X@W + bias + R), batched over heads ----------------
// One wave computes a 16 x (16*NT) tile of Y using V_WMMA_F32_16X16X4_F32.
// Out-of-range rows/cols are CLAMPED for loads (they only pollute output elements
// that are masked at store time), so the main K loop has no predication at all.
template <int NT>
__global__ void k_gemm(const float* __restrict__ X, const float* __restrict__ W,
                       const float* __restrict__ bias, const float* __restrict__ R,
                       float* __restrict__ Y,
                       int M, int N, int K, int ldx, int ldw, int ldy,
                       long sW, long sB, long sY, int act) {
    int head = blockIdx.y;
    W += (long)head * sW;
    if (bias) bias += (long)head * sB;
    Y += (long)head * sY;
    int wave = threadIdx.x >> 5;
    int lane = threadIdx.x & 31;
    int tiles_n = (N + 16 * NT - 1) / (16 * NT);
    int tiles_m = (M + 15) >> 4;
    int tile = blockIdx.x * (blockDim.x >> 5) + wave;
    if (tile >= tiles_m * tiles_n) return;
    int tm = tile / tiles_n, tn = tile % tiles_n;
    int half = lane >> 4;          // selects K pair {k0,k0+1} vs {k0+2,k0+3}
    int mr = lane & 15;
    int row = tm * 16 + mr;
    if (row > M - 1) row = M - 1;  // clamp (defensive; M%16==0 in all uses)
    const float* Xp = X + (long)row * ldx;

    int col[NT];
    bool cok[NT];
    const float* Wp[NT];
    for (int t = 0; t < NT; ++t) {
        int cc = tn * 16 * NT + t * 16 + mr;
        cok[t] = cc < N;
        col[t] = cc;
        int ccl = cok[t] ? cc : (N - 1);   // clamp for loads
        Wp[t] = W + ccl;
    }

    v8f acc[NT];
    for (int t = 0; t < NT; ++t) acc[t] = (v8f){};

    const int koff = 2 * half;
    const int K4 = K & ~3;
#pragma unroll 2
    for (int k0 = 0; k0 < K4; k0 += 4) {
        int ka = k0 + koff;
        v2f a;
        a.x = Xp[ka];
        a.y = Xp[ka + 1];
        for (int t = 0; t < NT; ++t) {
            v2f b;
            b.x = Wp[t][(long)ka * ldw];
            b.y = Wp[t][(long)(ka + 1) * ldw];
            acc[t] = __builtin_amdgcn_wmma_f32_16x16x4_f32(false, a, false, b, (short)0,
                                                           acc[t], false, false);
        }
    }
    if (K4 < K) {                       // single guarded tail iteration
        int ka = K4 + koff;
        bool g0 = ka < K, g1 = (ka + 1) < K;
        v2f a;
        a.x = g0 ? Xp[ka] : 0.f;
        a.y = g1 ? Xp[ka + 1] : 0.f;
        for (int t = 0; t < NT; ++t) {
            v2f b;
            b.x = g0 ? Wp[t][(long)ka * ldw] : 0.f;
            b.y = g1 ? Wp[t][(long)(ka + 1) * ldw] : 0.f;
            acc[t] = __builtin_amdgcn_wmma_f32_16x16x4_f32(false, a, false, b, (short)0,
                                                           acc[t], false, false);
        }
    }

    for (int t = 0; t < NT; ++t) {
        if (!cok[t]) continue;
        float bv = bias ? bias[col[t]] : 0.f;
        for (int r = 0; r < 8; ++r) {
            int m = tm * 16 + r + 8 * half;
            if (m < M) {
                float y = acc[t][r] + bv;
                if (R) y += R[(long)m * ldy + col[t]];
                if (act == 1) y = gelu_f(y);
                Y[(long)m * ldy + col[t]] = y;
            }
        }
    }
}

// ---------------- per-(head,batch) transformer attention, writes x2t layout ----------------
__global__ void k_attn(const float* __restrict__ Q, const float* __restrict__ Kb,
                       const float* __restrict__ V, float* __restrict__ X2T) {
    __shared__ float sq[124], sk[124], sv[310], att[62 * 62];
    int h = blockIdx.x / BSZ;
    int b = blockIdx.x % BSZ;
    const float* q = Q + ((long)h * BSZ + b) * 124;
    const float* k = Kb + ((long)h * BSZ + b) * 124;
    const float* v = V + ((long)h * BSZ + b) * 310;
    for (int i = threadIdx.x; i < 124; i += blockDim.x) { sq[i] = q[i]; sk[i] = k[i]; }
    for (int i = threadIdx.x; i < 310; i += blockDim.x) sv[i] = v[i];
    __syncthreads();
    const float sc = 0.70710678118654752f;
    if (threadIdx.x < 62) {
        int i = threadIdx.x;
        float q0 = sq[i * 2], q1 = sq[i * 2 + 1];
        float mx = -INFINITY;
        for (int j = 0; j < 62; ++j) {
            float s = (q0 * sk[j * 2] + q1 * sk[j * 2 + 1]) * sc;
            att[i * 62 + j] = s;
            mx = fmaxf(mx, s);
        }
        float sum = 0.f;
        for (int j = 0; j < 62; ++j) { float e = expf(att[i * 62 + j] - mx); att[i * 62 + j] = e; sum += e; }
        float r = 1.f / sum;
        for (int j = 0; j < 62; ++j) att[i * 62 + j] *= r;
    }
    __syncthreads();
    // o[i,c] = sum_j att[j,i] * v[j,c]  (torch-style transposed use)
    for (int idx = threadIdx.x; idx < 62 * 5; idx += blockDim.x) {
        int i = idx / 5, cc = idx % 5;
        float s = 0.f;
        for (int j = 0; j < 62; ++j) s += att[j * 62 + i] * sv[j * 5 + cc];
        X2T[(long)b * 1860 + i * 30 + h * 5 + cc] = s;
    }
}

// ---------------- concat xx = [xx1 (30) | xt (5)] along features ----------------
__global__ void k_catxx(const float* __restrict__ XX1, const float* __restrict__ XT,
                        float* __restrict__ XX) {
    int idx = blockIdx.x * blockDim.x + threadIdx.x;
    const int total = BSZ * 62 * 35;
    if (idx >= total) return;
    int f = idx % 35;
    int bn = idx / 35;
    float v;
    if (f < 30) v = XX1[(size_t)bn * 30 + f];
    else {
        int b = bn / 62, n = bn % 62;
        v = XT[(size_t)b * 310 + n * 5 + (f - 30)];
    }
    XX[idx] = v;
}

// ---------------- GAT attention: softmax over source axis i, out = relu(att @ h) ----------------
__global__ void k_gat_att(const float* __restrict__ H, int ldh, int coff,
                          const float* __restrict__ a, int Fo,
                          const float* __restrict__ Lmat,
                          float* __restrict__ Y, int ldy, int ycoff) {
    __shared__ float hl[62 * 32];
    __shared__ float s1[62], s2[62];
    __shared__ float att[62 * 62];
    int h = blockIdx.y;
    int b = blockIdx.x;
    const float* Hb = H + (long)b * 62 * ldh + (long)h * coff;
    const float* av = a + (long)h * 2 * Fo;
    for (int i = threadIdx.x; i < 62 * Fo; i += blockDim.x) {
        int n = i / Fo, o = i % Fo;
        hl[i] = Hb[(long)n * ldh + o];
    }
    __syncthreads();
    if (threadIdx.x < 62) {
        int n = threadIdx.x;
        float a1 = 0.f, a2 = 0.f;
        for (int o = 0; o < Fo; ++o) {
            float v = hl[n * Fo + o];
            float lv = v > 0.f ? v : 0.2f * v;
            a1 += lv * av[o];
            a2 += lv * av[Fo + o];
        }
        s1[n] = a1; s2[n] = a2;
    }
    __syncthreads();
    if (threadIdx.x < 62) {       // thread j handles column j; softmax over i
        int j = threadIdx.x;
        float mx = -INFINITY;
        for (int i = 0; i < 62; ++i) {
            float dj = 0.8f * Lmat[i * 62 + j] + 0.2f * (s1[i] + s2[j]);
            float m = dj > 0.f ? dj : -1e12f;
            att[i * 62 + j] = m;
            mx = fmaxf(mx, m);
        }
        float sum = 0.f;
        for (int i = 0; i < 62; ++i) { float e = expf(att[i * 62 + j] - mx); att[i * 62 + j] = e; sum += e; }
        float r = 1.f / sum;
        for (int i = 0; i < 62; ++i) att[i * 62 + j] *= r;
    }
    __syncthreads();
    float* Yb = Y + (long)b * 62 * ldy + (long)h * ycoff;
    for (int idx = threadIdx.x; idx < 62 * Fo; idx += blockDim.x) {
        int i = idx / Fo, o = idx % Fo;
        float s = 0.f;
        for (int j = 0; j < 62; ++j) s += att[i * 62 + j] * hl[j * Fo + o];
        Yb[(long)i * ldy + o] = fmaxf(s, 0.f);
    }
}

// ---------------- elu + log_softmax over 10, writes x11 region of d_out ----------------
__global__ void k_elu_ls(const float* __restrict__ G, float* __restrict__ X11) {
    int idx = blockIdx.x * blockDim.x + threadIdx.x;  // row = b*62+n
    if (idx >= BSZ * 62) return;
    const float* g = G + (size_t)idx * 10;
    float y[10];
    float mx = -INFINITY;
    for (int c = 0; c < 10; ++c) {
        float v = g[c];
        v = v > 0.f ? v : (expf(v) - 1.f);
        y[c] = v; mx = fmaxf(mx, v);
    }
    float s = 0.f;
    for (int c = 0; c < 10; ++c) s += expf(y[c] - mx);
    float ls = mx + logf(s);
    for (int c = 0; c < 10; ++c) X11[(size_t)idx * 10 + c] = y[c] - ls;
}

extern "C" void kernel_launch(void* const* d_in, const int* in_sizes, int n_in,
                              void* d_out, int out_size, void* d_ws, size_t ws_size,
                              hipStream_t stream) {
    const float* x    = (const float*)d_in[0];
    const float* A    = (const float*)d_in[1];
    const float* cw3  = (const float*)d_in[2];
    const float* cb3  = (const float*)d_in[3];
    const float* cw5  = (const float*)d_in[4];
    const float* cb5  = (const float*)d_in[5];
    const float* bng  = (const float*)d_in[6];
    const float* bnb  = (const float*)d_in[7];
    const float* caw1 = (const float*)d_in[8];
    const float* caw2 = (const float*)d_in[9];
    const float* qw   = (const float*)d_in[10];
    const float* qb   = (const float*)d_in[11];
    const float* kw   = (const float*)d_in[12];
    const float* kb   = (const float*)d_in[13];
    const float* vw   = (const float*)d_in[14];
    const float* vb   = (const float*)d_in[15];
    const float* lin1w= (const float*)d_in[16];
    const float* lin1b= (const float*)d_in[17];
    const float* ln1g = (const float*)d_in[18];
    const float* ln1bb= (const float*)d_in[19];
    const float* ln2g = (const float*)d_in[20];
    const float* ln2bb= (const float*)d_in[21];
    const float* fw1  = (const float*)d_in[22];
    const float* fb1  = (const float*)d_in[23];
    const float* fw2  = (const float*)d_in[24];
    const float* fb2  = (const float*)d_in[25];
    const float* gatw = (const float*)d_in[26];
    const float* gata = (const float*)d_in[27];
    const float* outw = (const float*)d_in[28];
    const float* outa = (const float*)d_in[29];
    const float* fc1w = (const float*)d_in[30];
    const float* fc1b = (const float*)d_in[31];
    const float* fc2w = (const float*)d_in[32];
    const float* fc2b = (const float*)d_in[33];
    const float* fc3w = (const float*)d_in[34];
    const float* fc3b = (const float*)d_in[35];

    float* ws = (float*)d_ws;
    size_t off = 0;
    auto take = [&](size_t n) { float* p = ws + off; off += n; return p; };

    float* Lm  = take(4096);
    size_t poolBase = off;
    float* x1b  = take((size_t)BSZ * 62 * 3);
    float* x2b  = take((size_t)BSZ * 62 * 8);
    float* x3b  = take((size_t)BSZ * 62 * 6);
    float* x5b  = take((size_t)BSZ * 62 * 12);
    float* x6b  = take((size_t)BSZ * 62 * 4);
    float* x8b  = take((size_t)BSZ * 62 * 8);
    float* x9b  = take((size_t)BSZ * 62 * 6);
    float* x11b = take((size_t)BSZ * 62 * 10);
    float* xx1b = take((size_t)BSZ * 62 * 30);
    float* xfb  = take((size_t)BSZ * 310);
    float* q6b  = take((size_t)6 * BSZ * 124);
    float* k6b  = take((size_t)6 * BSZ * 124);
    float* v6b  = take((size_t)6 * BSZ * 310);
    float* x2tb = take((size_t)BSZ * 1860);
    float* x21b = take((size_t)BSZ * 310);
    float* x31b = take((size_t)BSZ * 310);
    float* hgb  = take((size_t)BSZ * 186);
    float* xtb  = take((size_t)BSZ * 310);
    float* xxb  = take((size_t)BSZ * 62 * 35);
    // phase-B aliases (conv/transformer pool is dead once xx is built)
    float* hbufb = ws + poolBase;                                  // [B*62,128]
    float* xcatb = xxb;                                            // [B*62,128], overlays xx (dead after gat1 gemm)
    float* h2b   = ws + poolBase + (size_t)BSZ * 62 * 128;         // [B*62,10]
    float* outgb = h2b + (size_t)BSZ * 62 * 10;                    // [B,62,10]
    float* t1b   = ws + poolBase;                                  // [B,256] (hbuf dead by fc time)
    float* t2b   = t1b + (size_t)BSZ * 256;                        // [B,32]

    float* x11out = (float*)d_out;                      // [B,620]
    float* x4out  = (float*)d_out + (size_t)BSZ * 620;  // [B,4]

    auto gemm = [&](const float* X, int ldx, const float* W, int ldw, const float* bias,
                    const float* R, float* Y, int ldy, int M, int N, int K,
                    long sW, long sB, long sY, int heads, int act) {
        if (N > 16) {
            int tiles = ((M + 15) / 16) * ((N + 31) / 32);
            int blocks = (tiles + 7) / 8;
            k_gemm<2><<<dim3(blocks, heads), 256, 0, stream>>>(X, W, bias, R, Y, M, N, K,
                                                               ldx, ldw, ldy, sW, sB, sY, act);
        } else {
            int tiles = ((M + 15) / 16) * ((N + 15) / 16);
            int blocks = (tiles + 7) / 8;
            k_gemm<1><<<dim3(blocks, heads), 256, 0, stream>>>(X, W, bias, R, Y, M, N, K,
                                                               ldx, ldw, ldy, sW, sB, sY, act);
        }
    };

    // graph L
    k_normA<<<1, 64, 0, stream>>>(A, Lm);

    // ---- conv branch ----
    k_conv<<<BSZ, 256, 0, stream>>>(x, 5, nullptr, 0, cw3, cb3, bng, bnb, 3, x1b);
    k_chanatt<<<BSZ, 64, 0, stream>>>(x1b, 3, x, 5, nullptr, 0, caw1, caw2, x2b);
    k_conv<<<BSZ, 256, 0, stream>>>(x2b, 8, nullptr, 0, cw3 + 1 * 62 * 62 * 3, cb3 + 62,
                                    bng + 62, bnb + 62, 3, x3b);
    k_conv<<<BSZ, 256, 0, stream>>>(x2b, 8, x3b, 6, cw3 + 2 * 62 * 62 * 3, cb3 + 2 * 62,
                                    bng + 2 * 62, bnb + 2 * 62, 3, x5b);
    k_conv<<<BSZ, 256, 0, stream>>>(x2b, 8, nullptr, 0, cw5, cb5, bng + 3 * 62, bnb + 3 * 62, 5, x6b);
    k_conv<<<BSZ, 256, 0, stream>>>(x6b, 4, x2b, 8, cw5 + 1 * 62 * 62 * 5, cb5 + 62,
                                    bng + 4 * 62, bnb + 4 * 62, 5, x8b);
    k_conv<<<BSZ, 256, 0, stream>>>(x2b, 8, nullptr, 0, cw3 + 3 * 62 * 62 * 3, cb3 + 3 * 62,
                                    bng + 5 * 62, bnb + 5 * 62, 3, x9b);
    k_conv<<<BSZ, 256, 0, stream>>>(x9b, 6, x2b, 8, cw5 + 2 * 62 * 62 * 5, cb5 + 2 * 62,
                                    bng + 6 * 62, bnb + 6 * 62, 5, x11b);
    k_chanatt<<<BSZ, 64, 0, stream>>>(x5b, 12, x8b, 8, x11b, 10, caw1 + 15 * 62, caw2 + 62 * 15, xx1b);

    // ---- transformer branch ----
    k_ln<<<(BSZ * 62 + 255) / 256, 256, 0, stream>>>(x, ln1g, ln1bb, xfb, BSZ * 62);
    gemm(xfb, 310, qw, 124, qb, nullptr, q6b, 124, BSZ, 124, 310,
         (long)310 * 124, 124, (long)BSZ * 124, 6, 0);
    gemm(xfb, 310, kw, 124, kb, nullptr, k6b, 124, BSZ, 124, 310,
         (long)310 * 124, 124, (long)BSZ * 124, 6, 0);
    gemm(xfb, 310, vw, 310, vb, nullptr, v6b, 310, BSZ, 310, 310,
         (long)310 * 310, 310, (long)BSZ * 310, 6, 0);
    k_attn<<<6 * BSZ, 128, 0, stream>>>(q6b, k6b, v6b, x2tb);
    gemm(x2tb, 1860, lin1w, 310, lin1b, x, x21b, 310, BSZ, 310, 1860, 0, 0, 0, 1, 0);
    k_ln<<<(BSZ * 62 + 255) / 256, 256, 0, stream>>>(x21b, ln2g, ln2bb, x31b, BSZ * 62);
    gemm(x31b, 310, fw1, 186, fb1, nullptr, hgb, 186, BSZ, 186, 310, 0, 0, 0, 1, 1);  // gelu
    gemm(hgb, 186, fw2, 310, fb2, x21b, xtb, 310, BSZ, 310, 186, 0, 0, 0, 1, 0);

    // ---- GAT ----
    k_catxx<<<(BSZ * 62 * 35 + 255) / 256, 256, 0, stream>>>(xx1b, xtb, xxb);
    // h = xx @ gat_w per head -> hbuf [B*62,128] with per-head column blocks
    gemm(xxb, 35, gatw, 32, nullptr, nullptr, hbufb, 128, BSZ * 62, 32, 35,
         (long)35 * 32, 0, 32, 4, 0);
    k_gat_att<<<dim3(BSZ, 4), 128, 0, stream>>>(hbufb, 128, 32, gata, 32, Lm, xcatb, 128, 32);
    // out GAT: h2 = xcat @ out_w
    gemm(xcatb, 128, outw, 10, nullptr, nullptr, h2b, 10, BSZ * 62, 10, 128, 0, 0, 0, 1, 0);
    k_gat_att<<<dim3(BSZ, 1), 128, 0, stream>>>(h2b, 10, 0, outa, 10, Lm, outgb, 10, 0);
    k_elu_ls<<<(BSZ * 62 + 255) / 256, 256, 0, stream>>>(outgb, x11out);

    // ---- FC head ----
    gemm(x11out, 620, fc1w, 256, fc1b, nullptr, t1b, 256, BSZ, 256, 620, 0, 0, 0, 1, 0);
    gemm(t1b, 256, fc2w, 32, fc2b, nullptr, t2b, 32, BSZ, 32, 256, 0, 0, 0, 1, 0);
    gemm(t2b, 32, fc3w, 4, fc3b, nullptr, x4out, 4, BSZ, 4, 32, 0, 0, 0, 1, 0);

    (void)in_sizes; (void)n_in; (void)out_size; (void)ws_size;
}